// EdgeConv_81046032876027
// MI455X (gfx1250) — compile-verified
//
#include <hip/hip_runtime.h>

// EdgeConv (DGCNN layer) for MI455X / gfx1250, fp32 end-to-end.
// B=16, N=2048, F=64, filters=64, k=16.

#define BB   16
#define NN   2048
#define FF   64
#define FIL  64
#define KNN  16
#define EPSI 0.001f

typedef __attribute__((ext_vector_type(2))) float v2f;
typedef __attribute__((ext_vector_type(8))) float v8f;

__device__ __forceinline__ v8f wmma_f32(v2f a, v2f b, v8f c) {
  // D = A(16x4 f32) * B(4x16 f32) + C(16x16 f32)
  return __builtin_amdgcn_wmma_f32_16x16x4_f32(
      /*neg_a=*/false, a, /*neg_b=*/false, b,
      /*c_mod=*/(short)0, c, /*reuse_a=*/false, /*reuse_b=*/false);
}

// ---------------------------------------------------------------------------
// Kernel 1: per-point squared norm and "any nonzero" mask.
// ---------------------------------------------------------------------------
__global__ __launch_bounds__(256) void norms_kernel(const float* __restrict__ x,
                                                    float* __restrict__ xx,
                                                    float* __restrict__ maskf) {
  int i = blockIdx.x * blockDim.x + threadIdx.x;   // point id in [0, B*N)
  if (i >= BB * NN) return;
  const float* p = x + (size_t)i * FF;
  float s = 0.0f;
  float any = 0.0f;
#pragma unroll
  for (int c = 0; c < FF; ++c) {
    float v = p[c];
    s += v * v;
    if (v != 0.0f) any = 1.0f;
  }
  xx[i]    = s;
  maskf[i] = any;
}

// ---------------------------------------------------------------------------
// Kernel 2: fused distance GEMM (fp32 WMMA) + per-row top-16 selection.
// One workgroup = one batch b and one 16-row tile. 4 waves split the 128
// column tiles; per-wave top-16 lists are merged at the end.
// ---------------------------------------------------------------------------
#define W2V 4   // waves per workgroup
__global__ __launch_bounds__(32 * W2V) void knn_kernel(const float* __restrict__ x,
                                                       const float* __restrict__ xx,
                                                       const float* __restrict__ maskf,
                                                       int* __restrict__ idx) {
  __shared__ float tile[W2V][16][17];   // distance tile staging (padded)
  __shared__ float topd[W2V][16][KNN];  // per-wave, per-row candidate dists
  __shared__ int   topi[W2V][16][KNN];  // per-wave, per-row candidate indices

  const int wg   = blockIdx.x;          // b*(N/16) + rowtile
  const int b    = wg >> 7;             // N/16 = 128
  const int row0 = (wg & 127) << 4;
  const int tid  = threadIdx.x;
  const int wave = tid >> 5;
  const int lane = tid & 31;
  const int l15  = lane & 15;
  const int hi   = lane >> 4;           // 0: K=0,1 half | 1: K=2,3 half

  const float* xb  = x     + (size_t)b * NN * FF;
  const float* xxb = xx    + (size_t)b * NN;
  const float* mb  = maskf + (size_t)b * NN;

  // Preload A fragments for the whole K loop (F = 64 -> 16 steps of 4).
  // A layout: lane (M) holds K = hi*2 + {0,1} in the two components.
  v2f afrag[16];
  {
    const float* arow = xb + (size_t)(row0 + l15) * FF + hi * 2;
#pragma unroll
    for (int kk = 0; kk < 16; ++kk) {
      afrag[kk].x = arow[kk * 4 + 0];
      afrag[kk].y = arow[kk * 4 + 1];
    }
  }

  // Row-side norms / masks per accumulator component (C layout: VGPR r holds
  // M = r + 8*hi, N = l15).
  float xxA[8], mA[8];
#pragma unroll
  for (int r = 0; r < 8; ++r) {
    int m  = r + hi * 8;
    xxA[r] = xxb[row0 + m];
    mA[r]  = mb[row0 + m];
  }

  // Init per-wave top-k lists.
  if (lane < 16) {
#pragma unroll
    for (int j = 0; j < KNN; ++j) {
      topd[wave][lane][j] = __builtin_inff();
      topi[wave][lane][j] = 0;
    }
  }
  float curMax  = __builtin_inff();
  int   curSlot = 0;
  __syncthreads();

  // 128 column tiles, strided across the 4 waves (32 each -> uniform trips).
  for (int ct = wave; ct < NN / 16; ct += W2V) {
    const int col0 = ct << 4;

    v8f acc = {};
    const float* brow = xb + (size_t)(col0 + l15) * FF + hi * 2;
#pragma unroll
    for (int kk = 0; kk < 16; ++kk) {
      v2f bf;
      bf.x = brow[kk * 4 + 0];
      bf.y = brow[kk * 4 + 1];
      acc = wmma_f32(afrag[kk], bf, acc);
    }

    const float xxB = xxb[col0 + l15];
    const float mBv = mb[col0 + l15];

    // d = ||xi||^2 - 2 xi.xj + ||xj||^2 + (1 - mi*mj)*1e6 ; stage in LDS.
#pragma unroll
    for (int r = 0; r < 8; ++r) {
      float d = xxA[r] + xxB - 2.0f * acc[r] + (1.0f - mA[r] * mBv) * 1000000.0f;
      tile[wave][r + hi * 8][l15] = d;
    }
    __syncthreads();

    // Insertion into per-row top-16 (unsorted list, tracked max slot).
    if (lane < 16) {
      const int m = lane;
#pragma unroll 4
      for (int c = 0; c < 16; ++c) {
        float v = tile[wave][m][c];
        if (v < curMax) {
          topd[wave][m][curSlot] = v;
          topi[wave][m][curSlot] = col0 + c;
          float mx = topd[wave][m][0];
          int   ms = 0;
#pragma unroll
          for (int j = 1; j < KNN; ++j) {
            float t = topd[wave][m][j];
            if (t > mx) { mx = t; ms = j; }
          }
          curMax  = mx;
          curSlot = ms;
        }
      }
    }
    __syncthreads();
  }

  // Merge the 4 per-wave lists (64 candidates) -> global top-16 per row.
  if (tid < 16) {
    const int m = tid;
    float bd[KNN];
    int   bi[KNN];
#pragma unroll
    for (int j = 0; j < KNN; ++j) { bd[j] = __builtin_inff(); bi[j] = 0; }
    float mx = __builtin_inff();
    int   ms = 0;
    for (int w = 0; w < W2V; ++w) {
      for (int j = 0; j < KNN; ++j) {
        float v = topd[w][m][j];
        if (v < mx) {
          bd[ms] = v;
          bi[ms] = topi[w][m][j];
          mx = bd[0]; ms = 0;
          for (int t = 1; t < KNN; ++t)
            if (bd[t] > mx) { mx = bd[t]; ms = t; }
        }
      }
    }
    int* op = idx + ((size_t)b * NN + row0 + m) * KNN;
#pragma unroll
    for (int j = 0; j < KNN; ++j) op[j] = bi[j];
  }
}

// ---------------------------------------------------------------------------
// Kernel 3: edge features + MLP1 + BN + MLP2 + BN + max over k.
// One wave per point; 4 waves per workgroup. All GEMMs via fp32 WMMA.
// ---------------------------------------------------------------------------
#define W3V 4
__global__ __launch_bounds__(32 * W3V) void mlp_kernel(
    const float* __restrict__ x, const int* __restrict__ idx,
    const float* __restrict__ maskf,
    const float* __restrict__ W1, const float* __restrict__ b1,
    const float* __restrict__ g1, const float* __restrict__ be1,
    const float* __restrict__ mu1, const float* __restrict__ var1,
    const float* __restrict__ W2, const float* __restrict__ b2,
    const float* __restrict__ g2, const float* __restrict__ be2,
    const float* __restrict__ mu2, const float* __restrict__ var2,
    float* __restrict__ out) {
  __shared__ float edge[W3V][16][2 * FF];  // 16 x 128 per wave (32 KB total)
  __shared__ float hb[W3V][16][FIL];       // 16 x 64  per wave (16 KB total)

  const int tid  = threadIdx.x;
  const int wave = tid >> 5;
  const int lane = tid & 31;
  const int l15  = lane & 15;
  const int hi   = lane >> 4;

  const int p = blockIdx.x * W3V + wave;   // point id in [0, B*N)
  const int b = p >> 11;                   // N = 2048
  const int n = p & (NN - 1);

  const float* xb  = x + (size_t)b * NN * FF;
  const float  mk  = maskf[(size_t)b * NN + n];
  const float* cen = xb + (size_t)n * FF;

  // Build edge features: [central | neighbor*mask - central].
  {
    const int kk   = lane >> 1;
    const int half = lane & 1;
    const int nb   = idx[(size_t)p * KNN + kk];
    const float* nei = xb + (size_t)nb * FF;
    if (half == 0) {
#pragma unroll
      for (int c = 0; c < FF; ++c) edge[wave][kk][c] = cen[c];
    } else {
#pragma unroll
      for (int c = 0; c < FF; ++c) edge[wave][kk][FF + c] = nei[c] * mk - cen[c];
    }
  }
  __syncthreads();

  // ---- Layer 1: (16 x 128) * (128 x 64) ----
  v8f a0 = {}, a1 = {}, a2 = {}, a3 = {};
#pragma unroll 8
  for (int ks = 0; ks < 32; ++ks) {
    const int kc = ks * 4 + hi * 2;
    v2f a;
    a.x = edge[wave][l15][kc + 0];
    a.y = edge[wave][l15][kc + 1];
    const float* w0 = W1 + (size_t)kc * FIL;
    const float* w1 = W1 + (size_t)(kc + 1) * FIL;
    v2f bb;
    bb.x = w0[ 0 + l15]; bb.y = w1[ 0 + l15]; a0 = wmma_f32(a, bb, a0);
    bb.x = w0[16 + l15]; bb.y = w1[16 + l15]; a1 = wmma_f32(a, bb, a1);
    bb.x = w0[32 + l15]; bb.y = w1[32 + l15]; a2 = wmma_f32(a, bb, a2);
    bb.x = w0[48 + l15]; bb.y = w1[48 + l15]; a3 = wmma_f32(a, bb, a3);
  }

  // bias + ReLU + BN1, restage into LDS (C layout: VGPR r -> row r+8*hi, col l15).
#pragma unroll
  for (int nt = 0; nt < 4; ++nt) {
    const v8f acc = (nt == 0) ? a0 : (nt == 1) ? a1 : (nt == 2) ? a2 : a3;
    const int j   = nt * 16 + l15;
    const float sc = g1[j] * rsqrtf(var1[j] + EPSI);
    const float bi = b1[j], mm = mu1[j], bt = be1[j];
#pragma unroll
    for (int r = 0; r < 8; ++r) {
      float h = acc[r] + bi;
      h = h > 0.0f ? h : 0.0f;
      h = (h - mm) * sc + bt;
      hb[wave][r + hi * 8][j] = h;
    }
  }
  __syncthreads();

  // ---- Layer 2: (16 x 64) * (64 x 64) ----
  v8f c0 = {}, c1 = {}, c2 = {}, c3 = {};
#pragma unroll 8
  for (int ks = 0; ks < 16; ++ks) {
    const int kc = ks * 4 + hi * 2;
    v2f a;
    a.x = hb[wave][l15][kc + 0];
    a.y = hb[wave][l15][kc + 1];
    const float* w0 = W2 + (size_t)kc * FIL;
    const float* w1 = W2 + (size_t)(kc + 1) * FIL;
    v2f bb;
    bb.x = w0[ 0 + l15]; bb.y = w1[ 0 + l15]; c0 = wmma_f32(a, bb, c0);
    bb.x = w0[16 + l15]; bb.y = w1[16 + l15]; c1 = wmma_f32(a, bb, c1);
    bb.x = w0[32 + l15]; bb.y = w1[32 + l15]; c2 = wmma_f32(a, bb, c2);
    bb.x = w0[48 + l15]; bb.y = w1[48 + l15]; c3 = wmma_f32(a, bb, c3);
  }

  // bias + ReLU + BN2 + max over the 16 neighbors, then store.
#pragma unroll
  for (int nt = 0; nt < 4; ++nt) {
    const v8f acc = (nt == 0) ? c0 : (nt == 1) ? c1 : (nt == 2) ? c2 : c3;
    const int j   = nt * 16 + l15;
    const float sc = g2[j] * rsqrtf(var2[j] + EPSI);
    const float bi = b2[j], mm = mu2[j], bt = be2[j];
    float mx = -__builtin_inff();
#pragma unroll
    for (int r = 0; r < 8; ++r) {
      float h = acc[r] + bi;
      h = h > 0.0f ? h : 0.0f;
      h = (h - mm) * sc + bt;
      mx = fmaxf(mx, h);
    }
    // combine rows 0-7 (hi=0 lanes) with rows 8-15 (hi=1 lanes)
    float other = __shfl_xor(mx, 16, 32);
    mx = fmaxf(mx, other);
    if (hi == 0) out[(size_t)p * FIL + j] = mx;
  }
}

// ---------------------------------------------------------------------------
// Launch
// ---------------------------------------------------------------------------
extern "C" void kernel_launch(void* const* d_in, const int* in_sizes, int n_in,
                              void* d_out, int out_size, void* d_ws, size_t ws_size,
                              hipStream_t stream) {
  const float* x    = (const float*)d_in[0];
  const float* W1   = (const float*)d_in[1];
  const float* b1   = (const float*)d_in[2];
  const float* g1   = (const float*)d_in[3];
  const float* be1  = (const float*)d_in[4];
  const float* mu1  = (const float*)d_in[5];
  const float* var1 = (const float*)d_in[6];
  const float* W2   = (const float*)d_in[7];
  const float* b2   = (const float*)d_in[8];
  const float* g2   = (const float*)d_in[9];
  const float* be2  = (const float*)d_in[10];
  const float* mu2  = (const float*)d_in[11];
  const float* var2 = (const float*)d_in[12];
  float* out = (float*)d_out;

  // Workspace layout: xx (B*N f32) | mask (B*N f32) | idx (B*N*16 i32) ~2.25MB
  float* xx    = (float*)d_ws;
  float* maskf = xx + (size_t)BB * NN;
  int*   idx   = (int*)(maskf + (size_t)BB * NN);

  norms_kernel<<<(BB * NN) / 256, 256, 0, stream>>>(x, xx, maskf);
  knn_kernel<<<BB * (NN / 16), 32 * W2V, 0, stream>>>(x, xx, maskf, idx);
  mlp_kernel<<<(BB * NN) / W3V, 32 * W3V, 0, stream>>>(
      x, idx, maskf, W1, b1, g1, be1, mu1, var1, W2, b2, g2, be2, mu2, var2, out);
}